// GridStencilMap_68779606278319
// MI455X (gfx1250) — compile-verified
//
#include <hip/hip_runtime.h>
#include <stdint.h>

#define GRID_N    128
#define NUM_CELLS (GRID_N * GRID_N * GRID_N)
#define D_FEAT    8
#define INV_CELL  128.0f

// Native L2 fp32 atomic add (GLOBAL_ATOMIC_ADD_F32), no CAS loop.
__device__ __forceinline__ void atomic_add_g(float* p, float v) {
    unsafeAtomicAdd(p, v);
}

// ---------------------------------------------------------------------------
// Kernel 0: zero the grid accumulator (16.7M floats) with b128 stores.
// ---------------------------------------------------------------------------
__global__ void __launch_bounds__(256)
zero_grid_kernel(float4* __restrict__ grid4, int n4) {
    int i = blockIdx.x * blockDim.x + threadIdx.x;
    if (i < n4) grid4[i] = make_float4(0.f, 0.f, 0.f, 0.f);
}

// ---------------------------------------------------------------------------
// Kernel 1: P2G scatter. One thread per point; 64 fp32 atomics into the
// L2-resident grid (64 MB < 192 MB L2).
// ---------------------------------------------------------------------------
__global__ void __launch_bounds__(256)
p2g_kernel(const float* __restrict__ pos,
           const float4* __restrict__ feat4,
           float* __restrict__ grid,
           int n) {
    int i = blockIdx.x * blockDim.x + threadIdx.x;
    if (i >= n) return;

    float rx = pos[3 * i + 0] * INV_CELL;
    float ry = pos[3 * i + 1] * INV_CELL;
    float rz = pos[3 * i + 2] * INV_CELL;
    int bx = (int)rx, by = (int)ry, bz = (int)rz;   // pos in [0,1) -> nonneg
    float fx = rx - (float)bx, fy = ry - (float)by, fz = rz - (float)bz;
    float wx[2] = {1.f - fx, fx};
    float wy[2] = {1.f - fy, fy};
    float wz[2] = {1.f - fz, fz};

    float4 f0 = feat4[2 * i + 0];   // global_load_b128
    float4 f1 = feat4[2 * i + 1];

#pragma unroll
    for (int ox = 0; ox < 2; ++ox) {
        int nx = (bx + ox) & (GRID_N - 1);
#pragma unroll
        for (int oy = 0; oy < 2; ++oy) {
            int ny = (by + oy) & (GRID_N - 1);
            float wxy = wx[ox] * wy[oy];
#pragma unroll
            for (int oz = 0; oz < 2; ++oz) {
                int nz = (bz + oz) & (GRID_N - 1);
                float w = wxy * wz[oz];
                int hid = ((nx << 7) | ny) << 7 | nz;
                float* cell = grid + ((size_t)hid * D_FEAT);
                atomic_add_g(cell + 0, w * f0.x);
                atomic_add_g(cell + 1, w * f0.y);
                atomic_add_g(cell + 2, w * f0.z);
                atomic_add_g(cell + 3, w * f0.w);
                atomic_add_g(cell + 4, w * f1.x);
                atomic_add_g(cell + 5, w * f1.y);
                atomic_add_g(cell + 6, w * f1.z);
                atomic_add_g(cell + 7, w * f1.w);
            }
        }
    }
}

// ---------------------------------------------------------------------------
// Kernel 2: G2P gather. One thread per point; 16 b128 gathers (L2 hits),
// weighted accumulate, 2 b128 stores.
// ---------------------------------------------------------------------------
__global__ void __launch_bounds__(256)
g2p_kernel(const float* __restrict__ pos,
           const float4* __restrict__ grid4,
           float4* __restrict__ out4,
           int n) {
    int i = blockIdx.x * blockDim.x + threadIdx.x;
    if (i >= n) return;

    float rx = pos[3 * i + 0] * INV_CELL;
    float ry = pos[3 * i + 1] * INV_CELL;
    float rz = pos[3 * i + 2] * INV_CELL;
    int bx = (int)rx, by = (int)ry, bz = (int)rz;
    float fx = rx - (float)bx, fy = ry - (float)by, fz = rz - (float)bz;
    float wx[2] = {1.f - fx, fx};
    float wy[2] = {1.f - fy, fy};
    float wz[2] = {1.f - fz, fz};

    float4 a0 = make_float4(0.f, 0.f, 0.f, 0.f);
    float4 a1 = make_float4(0.f, 0.f, 0.f, 0.f);

#pragma unroll
    for (int ox = 0; ox < 2; ++ox) {
        int nx = (bx + ox) & (GRID_N - 1);
#pragma unroll
        for (int oy = 0; oy < 2; ++oy) {
            int ny = (by + oy) & (GRID_N - 1);
            float wxy = wx[ox] * wy[oy];
#pragma unroll
            for (int oz = 0; oz < 2; ++oz) {
                int nz = (bz + oz) & (GRID_N - 1);
                float w = wxy * wz[oz];
                int hid = ((nx << 7) | ny) << 7 | nz;
                float4 g0 = grid4[(size_t)hid * 2 + 0];  // global_load_b128
                float4 g1 = grid4[(size_t)hid * 2 + 1];
                a0.x = fmaf(w, g0.x, a0.x);
                a0.y = fmaf(w, g0.y, a0.y);
                a0.z = fmaf(w, g0.z, a0.z);
                a0.w = fmaf(w, g0.w, a0.w);
                a1.x = fmaf(w, g1.x, a1.x);
                a1.y = fmaf(w, g1.y, a1.y);
                a1.z = fmaf(w, g1.z, a1.z);
                a1.w = fmaf(w, g1.w, a1.w);
            }
        }
    }
    out4[2 * i + 0] = a0;   // global_store_b128
    out4[2 * i + 1] = a1;
}

// ---------------------------------------------------------------------------
// Host-side launcher (graph-capture safe: only kernel launches on `stream`).
// ---------------------------------------------------------------------------
extern "C" void kernel_launch(void* const* d_in, const int* in_sizes, int n_in,
                              void* d_out, int out_size, void* d_ws, size_t ws_size,
                              hipStream_t stream) {
    const float*  pos   = (const float*)d_in[0];   // [N,3] f32
    const float4* feat4 = (const float4*)d_in[1];  // [N,8] f32 as [N,2] float4
    float4*       out4  = (float4*)d_out;          // [N,8] f32
    float*        grid  = (float*)d_ws;            // [128^3, 8] f32 = 64 MB scratch
    float4*       grid4 = (float4*)d_ws;

    const int n = in_sizes[0] / 3;                 // N points

    // 1) zero grid accumulator
    const int n4 = NUM_CELLS * D_FEAT / 4;         // 4,194,304 float4s
    zero_grid_kernel<<<(n4 + 255) / 256, 256, 0, stream>>>(grid4, n4);

    // 2) P2G scatter-add (L2-resident atomics)
    p2g_kernel<<<(n + 255) / 256, 256, 0, stream>>>(pos, feat4, grid, n);

    // 3) G2P gather
    g2p_kernel<<<(n + 255) / 256, 256, 0, stream>>>(pos, grid4, out4, n);
}